// NGNN_GCNConv_29446295781899
// MI455X (gfx1250) — compile-verified
//
#include <hip/hip_runtime.h>
#include <hip/hip_bf16.h>

typedef __attribute__((ext_vector_type(2))) float v2f;
typedef __attribute__((ext_vector_type(8))) float v8f;

#define D 128
#define TILE_ROWS 16
#define LDS_STRIDE 136   // 136*4 = 544 B row stride: 16B-aligned float4 stores, staggered banks
#define WAVES_PER_BLOCK 4

// ---------------- utility kernels ----------------

__global__ void zero_kernel(float* __restrict__ p, long n) {
  long i = (long)blockIdx.x * blockDim.x + threadIdx.x;
  long stride = (long)gridDim.x * blockDim.x;
  for (; i < n; i += stride) p[i] = 0.0f;
}

__global__ void degree_kernel(const int* __restrict__ src, const int* __restrict__ dst,
                              float* __restrict__ deg_out, float* __restrict__ deg_in,
                              int n_edges) {
  int i = blockIdx.x * blockDim.x + threadIdx.x;
  if (i < n_edges) {
    unsafeAtomicAdd(&deg_out[src[i]], 1.0f);  // global_atomic_add_f32
    unsafeAtomicAdd(&deg_in[dst[i]], 1.0f);
  }
}

// Re-pack a 128x128 row-major weight matrix into WMMA B-fragment order:
// Wpk[((k0*8 + nt)*32 + lane)*2 + j] = W[k0*4 + 2*(lane>=16) + j][nt*16 + (lane&15)]
// so the GEMM inner loop loads one v2f per lane, contiguous across lanes.
__global__ void pack_w_kernel(const float* __restrict__ W, float* __restrict__ Wpk) {
  int i = blockIdx.x * blockDim.x + threadIdx.x;   // 0 .. 8191
  if (i >= 32 * 8 * 32) return;
  int lane = i & 31;
  int nt   = (i >> 5) & 7;
  int k0   = i >> 8;
  int kbase = k0 * 4 + (lane >> 4) * 2;
  int col   = nt * 16 + (lane & 15);
  Wpk[2 * i]     = W[kbase * D + col];
  Wpk[2 * i + 1] = W[(kbase + 1) * D + col];
}

// One wave32 per edge: lane L handles 4 consecutive features (32 lanes x float4 = 512B row).
__global__ void scatter_kernel(const float* __restrict__ x, const int* __restrict__ src,
                               const int* __restrict__ dst, const float* __restrict__ deg_out,
                               float* __restrict__ agg, int n_edges) {
  int gtid = blockIdx.x * blockDim.x + threadIdx.x;
  int e = gtid >> 5;
  int lane = gtid & 31;
  if (e >= n_edges) return;
  int s = src[e], d = dst[e];
  float ns = rsqrtf(fmaxf(deg_out[s], 1.0f));
  float4 v = *(const float4*)(x + (size_t)s * D + lane * 4);
  float* a = agg + (size_t)d * D + lane * 4;
  unsafeAtomicAdd(a + 0, v.x * ns);
  unsafeAtomicAdd(a + 1, v.y * ns);
  unsafeAtomicAdd(a + 2, v.z * ns);
  unsafeAtomicAdd(a + 3, v.w * ns);
}

// ---------------- fused 3-layer MLP (WMMA f32 16x16x4) ----------------

// One 16x16 output tile per (wave, nt); A fragments read from the wave's LDS tile.
// A layout (ISA 7.12.2, 32-bit A 16x4): lane<16 -> M=lane, K = 4k+{0,1};
//                                       lane>=16 -> M=lane-16, K = 4k+{2,3}.
__device__ __forceinline__ void gemm_tile(const float (*Tl)[LDS_STRIDE],
                                          const float* __restrict__ Wp,
                                          const float* __restrict__ bias,
                                          int lane, v8f acc[8]) {
  const int half = lane >> 4;
  const int lmod = lane & 15;
#pragma unroll
  for (int nt = 0; nt < 8; ++nt) {
    // C/D layout: every row of column N gets bias[N]
    float bv = bias[nt * 16 + lmod];
    acc[nt] = (v8f){bv, bv, bv, bv, bv, bv, bv, bv};
  }
  for (int k0 = 0; k0 < 32; ++k0) {
    v2f a = *(const v2f*)(&Tl[lmod][4 * k0 + 2 * half]);
#pragma unroll
    for (int nt = 0; nt < 8; ++nt) {
      v2f b = *(const v2f*)(Wp + ((size_t)(k0 * 8 + nt) * 32 + lane) * 2);
      acc[nt] = __builtin_amdgcn_wmma_f32_16x16x4_f32(
          /*neg_a=*/false, a, /*neg_b=*/false, b,
          /*c_mod=*/(short)0, acc[nt], /*reuse_a=*/false, /*reuse_b=*/false);
    }
  }
}

__global__ __launch_bounds__(128) void fused_mlp_kernel(
    const float* __restrict__ agg, const float* __restrict__ deg_in,
    const float* __restrict__ Wpk0, const float* __restrict__ bc,
    const float* __restrict__ Wpk1, const float* __restrict__ b1,
    const float* __restrict__ Wpk2, const float* __restrict__ b2,
    float* __restrict__ out, int n_tiles) {
  __shared__ float T[WAVES_PER_BLOCK][TILE_ROWS][LDS_STRIDE];
  const int wave = threadIdx.x >> 5;
  const int lane = threadIdx.x & 31;
  int tile = blockIdx.x * WAVES_PER_BLOCK + wave;
  if (tile >= n_tiles) tile = n_tiles - 1;   // duplicate work, identical values (benign)
  const int rowbase = tile * TILE_ROWS;
  float (*Tl)[LDS_STRIDE] = T[wave];
  const int half = lane >> 4;
  const int lmod = lane & 15;

  // Stage agg * norm_in into the wave-private LDS tile (coalesced float4 per row).
  for (int r = 0; r < TILE_ROWS; ++r) {
    int row = rowbase + r;
    float ni = rsqrtf(fmaxf(deg_in[row], 1.0f));
    float4 v = *(const float4*)(agg + (size_t)row * D + lane * 4);
    v.x *= ni; v.y *= ni; v.z *= ni; v.w *= ni;
    *(float4*)(&Tl[r][lane * 4]) = v;
  }
  asm volatile("s_wait_dscnt 0" ::: "memory");  // cross-lane LDS visibility within wave

  v8f acc[8];

  // Layer 1: relu(agg_norm @ Wc + bc) -> LDS
  gemm_tile(Tl, Wpk0, bc, lane, acc);
#pragma unroll
  for (int nt = 0; nt < 8; ++nt)
#pragma unroll
    for (int r = 0; r < 8; ++r)
      Tl[r + 8 * half][nt * 16 + lmod] = fmaxf(acc[nt][r], 0.0f);
  asm volatile("s_wait_dscnt 0" ::: "memory");

  // Layer 2: relu(h1 @ W1 + b1) -> LDS
  gemm_tile(Tl, Wpk1, b1, lane, acc);
#pragma unroll
  for (int nt = 0; nt < 8; ++nt)
#pragma unroll
    for (int r = 0; r < 8; ++r)
      Tl[r + 8 * half][nt * 16 + lmod] = fmaxf(acc[nt][r], 0.0f);
  asm volatile("s_wait_dscnt 0" ::: "memory");

  // Layer 3: h2 @ W2 + b2 -> global out
  gemm_tile(Tl, Wpk2, b2, lane, acc);
#pragma unroll
  for (int nt = 0; nt < 8; ++nt)
#pragma unroll
    for (int r = 0; r < 8; ++r)
      out[(size_t)(rowbase + r + 8 * half) * D + nt * 16 + lmod] = acc[nt][r];
}

// ---------------- launcher ----------------

extern "C" void kernel_launch(void* const* d_in, const int* in_sizes, int n_in,
                              void* d_out, int out_size, void* d_ws, size_t ws_size,
                              hipStream_t stream) {
  const float* x  = (const float*)d_in[0];
  const int*   src = (const int*)d_in[1];
  const int*   dst = (const int*)d_in[2];
  const float* Wc = (const float*)d_in[3];
  const float* bc = (const float*)d_in[4];
  const float* W1 = (const float*)d_in[5];
  const float* b1 = (const float*)d_in[6];
  const float* W2 = (const float*)d_in[7];
  const float* b2 = (const float*)d_in[8];
  float* out = (float*)d_out;

  const int n_nodes = in_sizes[0] / D;     // 100000
  const int n_edges = in_sizes[1];         // 1600000

  // Workspace layout (floats): deg_out[N] | deg_in[N] | agg[N*128] | Wpk0/1/2[16384 each]
  float* ws      = (float*)d_ws;
  float* deg_out = ws;
  float* deg_in  = ws + n_nodes;
  float* agg     = ws + 2 * (size_t)n_nodes;
  float* wpk0    = agg + (size_t)n_nodes * D;
  float* wpk1    = wpk0 + 32 * 8 * 32 * 2;
  float* wpk2    = wpk1 + 32 * 8 * 32 * 2;

  long zero_n = (long)n_nodes * (D + 2);   // degrees + agg
  zero_kernel<<<2048, 256, 0, stream>>>(ws, zero_n);

  degree_kernel<<<(n_edges + 255) / 256, 256, 0, stream>>>(src, dst, deg_out, deg_in, n_edges);

  pack_w_kernel<<<32, 256, 0, stream>>>(Wc, wpk0);
  pack_w_kernel<<<32, 256, 0, stream>>>(W1, wpk1);
  pack_w_kernel<<<32, 256, 0, stream>>>(W2, wpk2);

  // one wave per edge
  int scatter_threads = n_edges * 32;
  scatter_kernel<<<(scatter_threads + 255) / 256, 256, 0, stream>>>(x, src, dst, deg_out, agg, n_edges);

  int n_tiles = (n_nodes + TILE_ROWS - 1) / TILE_ROWS;   // 6250
  fused_mlp_kernel<<<(n_tiles + WAVES_PER_BLOCK - 1) / WAVES_PER_BLOCK, 128, 0, stream>>>(
      agg, deg_in, wpk0, bc, wpk1, b1, wpk2, b2, out, n_tiles);
}